// DirectDistanceModel_5179730559253
// MI455X (gfx1250) — compile-verified
//
#include <hip/hip_runtime.h>
#include <hip/hip_bf16.h>
#include <stdint.h>

// ---------------- problem constants (fixed in the reference file) ------------
#define NITEMS   2000
#define NSTORAGE 4094
#define NLOCS    4096          // NSTORAGE + 2
#define NEDGES   4194304       // E
#define HID      32

// workspace layout (element counts)
#define WLOC_N   ((size_t)NLOCS * NLOCS)       // 16,777,216 u32
#define WSEQ_N   ((size_t)NITEMS * NITEMS)     //  4,000,000 u32
#define WITL_N   ((size_t)NITEMS)              //      2,000 u32
#define NBLK_II  2048                          // item-item reduce blocks

// byte offsets into d_ws (all 16B aligned)
#define OFF_WLOC 0
#define OFF_WSEQ (OFF_WLOC + WLOC_N * 4)
#define OFF_WITL (OFF_WSEQ + WSEQ_N * 4)
#define OFF_ITL  (OFF_WITL + WITL_N * 4)
#define OFF_BP   (OFF_ITL  + WITL_N * 4)
#define OFF_CMP  (OFF_BP   + NBLK_II * 4)

typedef __attribute__((ext_vector_type(2))) float v2f;
typedef __attribute__((ext_vector_type(8))) float v8f;

// ---------------- K0: zero winner arrays (uint4 stores) ----------------------
__global__ __launch_bounds__(256) void k_zero(uint4* __restrict__ p, long n4) {
    long stride = (long)gridDim.x * blockDim.x;
    for (long i = (long)blockIdx.x * blockDim.x + threadIdx.x; i < n4; i += stride)
        p[i] = make_uint4(0u, 0u, 0u, 0u);
}

// ---------------- K1: scatter edge ids (highest edge id wins) ----------------
__global__ __launch_bounds__(256) void k_scatter(
    const int* __restrict__ src, const int* __restrict__ dst,
    const uint8_t* __restrict__ mask,            // [E,3] bool bytes
    unsigned* __restrict__ wloc, unsigned* __restrict__ wseq,
    unsigned* __restrict__ witl) {
    int e = blockIdx.x * blockDim.x + threadIdx.x;
    if (e >= NEDGES) return;
    int s = src[e], d = dst[e];
    const uint8_t* m = mask + 3 * (size_t)e;
    unsigned tag = (unsigned)(e + 1);
    if (m[0]) {                                  // loc-loc
        int ls = s - NITEMS, ld = d - NITEMS;
        if (ls >= 0 && ls < NLOCS && ld >= 0 && ld < NLOCS)
            atomicMax(&wloc[(size_t)ls * NLOCS + ld], tag);
    }
    if (m[1]) {                                  // item-item
        if (s >= 0 && s < NITEMS && d >= 0 && d < NITEMS)
            atomicMax(&wseq[(size_t)s * NITEMS + d], tag);
    }
    if (m[2]) {                                  // item-loc
        int li = d - NITEMS;
        if (s >= 0 && s < NITEMS && li >= 0 && li < NSTORAGE)
            atomicMax(&witl[s], tag);
    }
}

// ---------------- K2: resolve item_to_loc ------------------------------------
__global__ __launch_bounds__(256) void k_resolve(
    const unsigned* __restrict__ witl, const int* __restrict__ dst,
    int* __restrict__ itl) {
    int i = blockIdx.x * blockDim.x + threadIdx.x;
    if (i >= NITEMS) return;
    unsigned w = witl[i];
    itl[i] = w ? (dst[w - 1] - NITEMS) : 0;      // matches where(v, li, 0) w/ zero init
}

// ---------------- K3: item_item_dist partial sums ----------------------------
__global__ __launch_bounds__(256) void k_item_item(
    const unsigned* __restrict__ wseq, const unsigned* __restrict__ wloc,
    const int* __restrict__ itl, const float* __restrict__ attr,
    float* __restrict__ bp) {
    __shared__ float red[256];
    float acc = 0.f;
    int stride = gridDim.x * blockDim.x;
    for (int idx = blockIdx.x * blockDim.x + threadIdx.x;
         idx < NITEMS * NITEMS; idx += stride) {
        unsigned w = wseq[idx];
        if (w) {
            int i = idx / NITEMS, j = idx - i * NITEMS;   // const div -> mulhi
            float sv = attr[2 * (size_t)(w - 1) + 1];
            unsigned lw = wloc[((size_t)itl[i] << 12) + itl[j]];
            if (lw) acc += sv * attr[2 * (size_t)(lw - 1)];
        }
    }
    int t = threadIdx.x;
    red[t] = acc; __syncthreads();
    for (int off = 128; off > 0; off >>= 1) {
        if (t < off) red[t] += red[t + off];
        __syncthreads();
    }
    if (t == 0) bp[blockIdx.x] = red[0];
}

// ---------------- K4: fold partials + start/end depot sums -> comps ----------
__global__ __launch_bounds__(256) void k_finalize(
    const unsigned* __restrict__ wloc, const float* __restrict__ attr,
    const int* __restrict__ itl, const float* __restrict__ bp,
    float* __restrict__ comps) {
    __shared__ float sA[256], sB[256], sC[256];
    int t = threadIdx.x;
    float a = 0.f, b = 0.f, c = 0.f;
    for (int k = t; k < NBLK_II; k += 256) a += bp[k];        // fixed order
    for (int i = t; i < NITEMS; i += 256) {
        int li = itl[i];
        unsigned ws = wloc[(size_t)NSTORAGE * NLOCS + li];    // start row 4094
        if (ws) b += attr[2 * (size_t)(ws - 1)];
        unsigned we = wloc[((size_t)li << 12) + (NSTORAGE + 1)]; // end col 4095
        if (we) c += attr[2 * (size_t)(we - 1)];
    }
    sA[t] = a; sB[t] = b; sC[t] = c; __syncthreads();
    for (int off = 128; off > 0; off >>= 1) {
        if (t < off) { sA[t] += sA[t+off]; sB[t] += sB[t+off]; sC[t] += sC[t+off]; }
        __syncthreads();
    }
    if (t == 0) { comps[0] = sA[0]; comps[1] = sB[0]; comps[2] = sC[0]; }
}

// ---------------- K5: tiny MLP via V_WMMA_F32_16X16X4_F32 --------------------
// h = relu([1,3]comps @ [3,32]W1 + b1);  pred = h @ [32,1]W2 + b2.
// A (16x4 f32, 2 VGPR/lane): lanes 0-15 hold M=lane, K={0,1}; lanes 16-31 K={2,3}.
//   Only row M=0 nonzero: lane0 = {c0,c1}, lane16 = {c2,0}.
// B (4x16 f32): row K striped across lanes in one VGPR: lanes 0-15 -> {K0,K1},
//   lanes 16-31 -> {K2,K3=0}. Two tiles cover W1 columns 0-15 / 16-31.
// D (16x16 f32, 8 VGPR): element 0, lanes 0-15 = row M=0, cols 0-15.
__global__ __launch_bounds__(32) void k_mlp(
    const float* __restrict__ comps, const float* __restrict__ W1,
    const float* __restrict__ bias1, const float* __restrict__ W2,
    const float* __restrict__ bias2, float* __restrict__ out) {
    int lane = threadIdx.x;
    int n = lane & 15;

    v2f a = {0.f, 0.f};
    if (lane == 0)  { a.x = comps[0]; a.y = comps[1]; }
    if (lane == 16) { a.x = comps[2]; a.y = 0.f; }

    v2f bt0, bt1;
    if (lane < 16) {
        bt0.x = W1[0 * HID + n];      bt0.y = W1[1 * HID + n];
        bt1.x = W1[0 * HID + 16 + n]; bt1.y = W1[1 * HID + 16 + n];
    } else {
        bt0.x = W1[2 * HID + n];      bt0.y = 0.f;
        bt1.x = W1[2 * HID + 16 + n]; bt1.y = 0.f;
    }

    v8f c = {};
    v8f d0 = __builtin_amdgcn_wmma_f32_16x16x4_f32(false, a, false, bt0,
                                                   (short)0, c, false, false);
    v8f d1 = __builtin_amdgcn_wmma_f32_16x16x4_f32(false, a, false, bt1,
                                                   (short)0, c, false, false);

    float partial = 0.f;
    if (lane < 16) {
        float h0 = d0[0] + bias1[n];      h0 = h0 > 0.f ? h0 : 0.f;
        float h1 = d1[0] + bias1[16 + n]; h1 = h1 > 0.f ? h1 : 0.f;
        partial = h0 * W2[n] + h1 * W2[16 + n];
    }
    __shared__ float red[32];
    red[lane] = partial;
    __syncthreads();
    if (lane == 0) {
        float s = bias2[0];
        for (int t = 0; t < 16; ++t) s += red[t];   // fixed order
        *out = s;
    }
}

// ---------------- launcher ---------------------------------------------------
extern "C" void kernel_launch(void* const* d_in, const int* in_sizes, int n_in,
                              void* d_out, int out_size, void* d_ws, size_t ws_size,
                              hipStream_t stream) {
    (void)in_sizes; (void)n_in; (void)out_size; (void)ws_size;
    const int*     ei   = (const int*)d_in[0];       // [2, E] int32
    const int*     src  = ei;
    const int*     dst  = ei + NEDGES;
    const float*   attr = (const float*)d_in[1];     // [E, 2]
    const uint8_t* mask = (const uint8_t*)d_in[2];   // [E, 3] bool bytes
    const float*   W1   = (const float*)d_in[6];     // [3, 32]
    const float*   b1   = (const float*)d_in[7];     // [32]
    const float*   W2   = (const float*)d_in[8];     // [32, 1]
    const float*   b2   = (const float*)d_in[9];     // [1]

    char* ws = (char*)d_ws;
    unsigned* wloc  = (unsigned*)(ws + OFF_WLOC);
    unsigned* wseq  = (unsigned*)(ws + OFF_WSEQ);
    unsigned* witl  = (unsigned*)(ws + OFF_WITL);
    int*      itl   = (int*)     (ws + OFF_ITL);
    float*    bp    = (float*)   (ws + OFF_BP);
    float*    comps = (float*)   (ws + OFF_CMP);

    // K0: zero winner arrays (wloc|wseq|witl are contiguous; count % 4 == 0)
    long n4 = (long)(WLOC_N + WSEQ_N + WITL_N) / 4;
    k_zero<<<4096, 256, 0, stream>>>((uint4*)wloc, n4);

    // K1: scatter (highest edge id wins -> deterministic)
    k_scatter<<<NEDGES / 256, 256, 0, stream>>>(src, dst, mask, wloc, wseq, witl);

    // K2: resolve item_to_loc
    k_resolve<<<(NITEMS + 255) / 256, 256, 0, stream>>>(witl, dst, itl);

    // K3: item-item masked gather-reduce (L2-resident gathers)
    k_item_item<<<NBLK_II, 256, 0, stream>>>(wseq, wloc, itl, attr, bp);

    // K4: fold partials + depot sums
    k_finalize<<<1, 256, 0, stream>>>(wloc, attr, itl, bp, comps);

    // K5: WMMA MLP head
    k_mlp<<<1, 32, 0, stream>>>(comps, W1, b1, W2, b2, (float*)d_out);
}